// GCN_70755291235030
// MI455X (gfx1250) — compile-verified
//
#include <hip/hip_runtime.h>
#include <math.h>

#define N_NODES 100000
#define IN_DIM  128
#define HIDDEN  128
#define LATENT  32

typedef float v2f __attribute__((ext_vector_type(2)));
typedef float v8f __attribute__((ext_vector_type(8)));

// ---------------------------------------------------------------------------
// CSR rowptr from row-sorted edge list (edges_proc is sorted by row because
// np.unique sorts enc = row*n + dst ascending). Every node has a self loop,
// so every node appears; the fill loop handles boundaries generically anyway.
// ---------------------------------------------------------------------------
__global__ void build_rowptr_kernel(const int* __restrict__ row, int E2, int n,
                                    int* __restrict__ rowptr) {
    int e = blockIdx.x * blockDim.x + threadIdx.x;
    if (e >= E2) return;
    int r  = row[e];
    int rp = (e == 0) ? -1 : row[e - 1];
    for (int j = rp + 1; j <= r; ++j) rowptr[j] = e;
    if (e == E2 - 1) {
        for (int j = r + 1; j <= n; ++j) rowptr[j] = E2;
    }
}

// deg[i] == #edges with col==i == #edges with row==i (symmetric edge set),
// i.e. the rowptr run length. dinv = deg^{-1/2}.
__global__ void dinv_kernel(const int* __restrict__ rowptr,
                            float* __restrict__ dinv, int n) {
    int i = blockIdx.x * blockDim.x + threadIdx.x;
    if (i < n) {
        float d = (float)(rowptr[i + 1] - rowptr[i]);
        dinv[i] = (d > 0.0f) ? rsqrtf(fmaxf(d, 1e-12f)) : 0.0f;
    }
}

__global__ void zero_f32(float* __restrict__ p, long n) {
    long i = (long)blockIdx.x * blockDim.x + threadIdx.x;
    if (i < n) p[i] = 0.0f;
}

// ---------------------------------------------------------------------------
// FP32 WMMA GEMM:  C[M,F] (+)= A[M,K] @ W[K,F]  [+ bias] [+ relu]
// One wave32 computes one 16x16 tile of C using V_WMMA_F32_16X16X4_F32.
// ISA 7.12.2 layouts:
//   A 16x4 f32 : lane L, m = L&15; lanes<16 hold K=k,k+1 ; lanes>=16 K=k+2,k+3
//   B 4x16 f32 : lane L, n = L&15; lanes<16 hold K=k,k+1 ; lanes>=16 K=k+2,k+3
//   C 16x16    : lane L, col n = L&15; VGPR j -> row j + (L>=16 ? 8 : 0)
// M,F multiples of 16 and K multiple of 4 for every call here, so EXEC stays
// all-ones inside each active wave (WMMA requirement).
// ---------------------------------------------------------------------------
__global__ __launch_bounds__(256) void gemm_wmma_f32(
    const float* __restrict__ A, const float* __restrict__ W,
    float* __restrict__ C, int M, int K, int F, int accum,
    const float* __restrict__ bias, int relu) {
    const int lane = threadIdx.x & 31;
    const int wib  = threadIdx.x >> 5;
    const int tiles_n = F >> 4;
    const int tiles_m = M >> 4;
    int wid = blockIdx.x * 8 + wib;
    int tm = wid / tiles_n;
    int tn = wid - tm * tiles_n;
    if (tm >= tiles_m) return;              // wave-uniform exit

    const int m0  = tm << 4;
    const int n0  = tn << 4;
    const int hi  = lane >> 4;              // lane half
    const int l15 = lane & 15;
    const int ka  = hi << 1;                // +0 or +2 K offset for this half

    v8f c;
    if (accum) {
#pragma unroll
        for (int j = 0; j < 8; ++j)
            c[j] = C[(size_t)(m0 + j + hi * 8) * F + n0 + l15];
    } else {
#pragma unroll
        for (int j = 0; j < 8; ++j) c[j] = 0.0f;
    }

    const float* Arow = A + (size_t)(m0 + l15) * K;
    for (int k = 0; k < K; k += 4) {
        v2f a, b;
        a.x = Arow[k + ka];
        a.y = Arow[k + ka + 1];
        b.x = W[(size_t)(k + ka)     * F + n0 + l15];
        b.y = W[(size_t)(k + ka + 1) * F + n0 + l15];
        // 8 args: (neg_a, A, neg_b, B, c_mod, C, reuse_a, reuse_b)
        c = __builtin_amdgcn_wmma_f32_16x16x4_f32(
                false, a, false, b, (short)0, c, false, false);
    }

    if (bias) {
        float bv = bias[n0 + l15];
#pragma unroll
        for (int j = 0; j < 8; ++j) c[j] += bv;
    }
    if (relu) {
#pragma unroll
        for (int j = 0; j < 8; ++j) c[j] = fmaxf(c[j], 0.0f);
    }

#pragma unroll
    for (int j = 0; j < 8; ++j)
        C[(size_t)(m0 + j + hi * 8) * F + n0 + l15] = c[j];
}

// ---------------------------------------------------------------------------
// GCN aggregation as a per-row gather (no atomics):
//   out[i,:] = relu( bias + sum_{e in [rowptr[i],rowptr[i+1])}
//                            dinv[i]*dinv[col[e]] * tmp[col[e],:] )
// One wave per node; lane holds 4 of the 128 features (float4), so each
// neighbor row is one coalesced 512B read. Equivalent to the reference's
// scatter over col by edge-set + norm symmetry.
// ---------------------------------------------------------------------------
__global__ __launch_bounds__(256) void gcn_aggregate_kernel(
    const float* __restrict__ tmp, const int* __restrict__ col,
    const int* __restrict__ rowptr, const float* __restrict__ dinv,
    const float* __restrict__ bias, float* __restrict__ out, int n) {
    int wid  = blockIdx.x * 8 + (threadIdx.x >> 5);
    int lane = threadIdx.x & 31;
    if (wid >= n) return;                   // wave-uniform
    int beg = rowptr[wid];
    int end = rowptr[wid + 1];
    float di = dinv[wid];
    float4 acc = make_float4(0.f, 0.f, 0.f, 0.f);
    for (int e = beg; e < end; ++e) {
        int c = col[e];
        if (e + 2 < end) {
            // warm L2/L0 for the row two neighbors ahead (global_prefetch_b8)
            __builtin_prefetch(tmp + (size_t)col[e + 2] * HIDDEN + lane * 4, 0, 1);
        }
        float w = di * dinv[c];
        float4 v = ((const float4*)(tmp + (size_t)c * HIDDEN))[lane];
        acc.x += w * v.x;
        acc.y += w * v.y;
        acc.z += w * v.z;
        acc.w += w * v.w;
    }
    float4 b4 = ((const float4*)bias)[lane];
    acc.x = fmaxf(acc.x + b4.x, 0.0f);
    acc.y = fmaxf(acc.y + b4.y, 0.0f);
    acc.z = fmaxf(acc.z + b4.z, 0.0f);
    acc.w = fmaxf(acc.w + b4.w, 0.0f);
    ((float4*)(out + (size_t)wid * HIDDEN))[lane] = acc;
}

// ---------------------------------------------------------------------------
// Link-prediction loss: one wave per edge, lane j holds latent dim j (32).
// softplus(sign * ((z[a]*z[b]) . Pw + Pb)), block-reduced via LDS, one
// global atomic per block.
// ---------------------------------------------------------------------------
__global__ __launch_bounds__(256) void edge_loss_kernel(
    const float* __restrict__ z,
    const int* __restrict__ ea, const int* __restrict__ eb,
    const float* __restrict__ Pw, const float* __restrict__ Pb,
    int E, float sign, float* __restrict__ acc) {
    __shared__ float red[8];
    long t = (long)blockIdx.x * blockDim.x + threadIdx.x;
    int e    = (int)(t >> 5);
    int lane = (int)(t & 31);
    int wib  = threadIdx.x >> 5;
    float l = 0.0f;
    if (e < E) {
        int a = ea[e];
        int b = eb[e];
        float p = z[(size_t)a * LATENT + lane] *
                  z[(size_t)b * LATENT + lane] * Pw[lane];
#pragma unroll
        for (int off = 16; off > 0; off >>= 1)
            p += __shfl_xor(p, off, 32);
        if (lane == 0) {
            float xv = sign * (p + Pb[0]);
            // numerically-stable softplus
            l = fmaxf(xv, 0.0f) + log1pf(expf(-fabsf(xv)));
        }
    }
    if (lane == 0) red[wib] = l;
    __syncthreads();
    if (threadIdx.x == 0) {
        float s = 0.0f;
#pragma unroll
        for (int i = 0; i < 8; ++i) s += red[i];
        atomicAdd(acc, s);
    }
}

__global__ void finalize_kernel(const float* __restrict__ acc,
                                float* __restrict__ out, float invE) {
    out[0] = acc[0] * invE + acc[1] * invE;
}

// ---------------------------------------------------------------------------
// Launch
// ---------------------------------------------------------------------------
extern "C" void kernel_launch(void* const* d_in, const int* in_sizes, int n_in,
                              void* d_out, int out_size, void* d_ws, size_t ws_size,
                              hipStream_t stream) {
    const float* x      = (const float*)d_in[0];
    const int*   edges  = (const int*)  d_in[1];
    const int*   neg    = (const int*)  d_in[2];
    const int*   eproc  = (const int*)  d_in[3];
    const float* W1 = (const float*)d_in[4];  const float* b1 = (const float*)d_in[5];
    const float* W2 = (const float*)d_in[6];  const float* b2 = (const float*)d_in[7];
    const float* W3 = (const float*)d_in[8];  const float* b3 = (const float*)d_in[9];
    const float* Lw1 = (const float*)d_in[10]; const float* Lb1 = (const float*)d_in[11];
    const float* Lw2 = (const float*)d_in[12]; const float* Lb2 = (const float*)d_in[13];
    const float* Pw  = (const float*)d_in[14]; const float* Pb  = (const float*)d_in[15];

    const int N  = N_NODES;
    const int E  = in_sizes[1] / 2;   // 800000
    const int E2 = in_sizes[3] / 2;   // processed (undirected + self loop) edges

    const int* row = eproc;
    const int* col = eproc + E2;

    // workspace layout
    int*   rowptr = (int*)d_ws;                       // N+1 ints
    float* dinv   = (float*)(rowptr + (N + 2));       // N floats (keep 8B align)
    float* tmp    = dinv + N;                         // N*HIDDEN
    float* x1     = tmp  + (size_t)N * HIDDEN;
    float* x2     = x1   + (size_t)N * HIDDEN;
    float* x3     = x2   + (size_t)N * HIDDEN;
    float* z      = x3   + (size_t)N * HIDDEN;        // N*LATENT
    float* acc    = z    + (size_t)N * LATENT;        // 2

    dim3 blk(256);

    auto gemm = [&](const float* A, const float* W, float* C,
                    int M, int K, int F, int accum,
                    const float* bias, int relu) {
        long tiles = (long)(M / 16) * (F / 16);
        gemm_wmma_f32<<<dim3((unsigned)((tiles + 7) / 8)), blk, 0, stream>>>(
            A, W, C, M, K, F, accum, bias, relu);
    };
    auto aggregate = [&](const float* src, const float* bias, float* dst) {
        gcn_aggregate_kernel<<<dim3((N + 7) / 8), blk, 0, stream>>>(
            src, col, rowptr, dinv, bias, dst, N);
    };

    // --- CSR index + symmetric normalization (no atomics) ---
    build_rowptr_kernel<<<dim3((E2 + 255) / 256), blk, 0, stream>>>(row, E2, N, rowptr);
    dinv_kernel<<<dim3((N + 255) / 256), blk, 0, stream>>>(rowptr, dinv, N);

    // --- conv1: x1 = relu(aggregate(x @ W1) + b1) ---
    gemm(x, W1, tmp, N, IN_DIM, HIDDEN, 0, nullptr, 0);
    aggregate(tmp, b1, x1);

    // --- conv2: x2 = relu(aggregate(x1 @ W2) + b2) ---
    gemm(x1, W2, tmp, N, HIDDEN, HIDDEN, 0, nullptr, 0);
    aggregate(tmp, b2, x2);

    // --- conv3: x3 = relu(aggregate(x1 @ W3) + b3) ---
    gemm(x1, W3, tmp, N, HIDDEN, HIDDEN, 0, nullptr, 0);
    aggregate(tmp, b3, x3);

    // --- JKNet: h = relu([x1|x2|x3] @ Lw1 + Lb1) via 3 accumulating GEMMs,
    //     bias+relu fused into the last one ---
    gemm(x1, Lw1,                tmp, N, HIDDEN, HIDDEN, 0, nullptr, 0);
    gemm(x2, Lw1 + 128 * HIDDEN, tmp, N, HIDDEN, HIDDEN, 1, nullptr, 0);
    gemm(x3, Lw1 + 256 * HIDDEN, tmp, N, HIDDEN, HIDDEN, 1, Lb1, 1);

    // --- z = relu(h @ Lw2 + Lb2), epilogue fused ---
    gemm(tmp, Lw2, z, N, HIDDEN, LATENT, 0, Lb2, 1);

    // --- link prediction loss ---
    zero_f32<<<dim3(1), dim3(64), 0, stream>>>(acc, 2);
    {
        unsigned g = (unsigned)(((long)E + 7) / 8);
        edge_loss_kernel<<<dim3(g), blk, 0, stream>>>(
            z, edges, edges + E, Pw, Pb, E, -1.0f, acc);       // pos: softplus(-l)
        edge_loss_kernel<<<dim3(g), blk, 0, stream>>>(
            z, neg, neg + E, Pw, Pb, E, 1.0f, acc + 1);        // neg: softplus(l)
    }
    finalize_kernel<<<dim3(1), dim3(1), 0, stream>>>(acc, (float*)d_out, 1.0f / (float)E);
}